// KalmanFilter_592705487346
// MI455X (gfx1250) — compile-verified
//
#include <hip/hip_runtime.h>

typedef __attribute__((ext_vector_type(2))) float v2f;
typedef __attribute__((ext_vector_type(8))) float v8f;

// ---------------------------------------------------------------------------
// Setup: one thread computes the shared Kalman algebra, emits
//   - padded combined transform M6 -> d_ws as float M[16][8] (zero padded)
//   - H_update (16 floats) -> d_out + 4*Bc
// ---------------------------------------------------------------------------
__global__ void kalman_setup(const float* __restrict__ Nmat,
                             const float* __restrict__ H,
                             const float* __restrict__ phi,
                             const float* __restrict__ psi,
                             float* __restrict__ M,      // ws: [16][8]
                             float* __restrict__ Hout) { // out + 4*Bc
    if (threadIdx.x != 0 || blockIdx.x != 0) return;

    float tmp[4][4], Hp[4][4];
    // tmp = phi @ H ; Hp = tmp @ phi^T
    for (int i = 0; i < 4; ++i)
        for (int j = 0; j < 4; ++j) {
            float s = 0.f;
            for (int k = 0; k < 4; ++k) s += phi[i * 4 + k] * H[k * 4 + j];
            tmp[i][j] = s;
        }
    for (int i = 0; i < 4; ++i)
        for (int j = 0; j < 4; ++j) {
            float s = 0.f;
            for (int k = 0; k < 4; ++k) s += tmp[i][k] * phi[j * 4 + k];
            Hp[i][j] = s;
        }
    // HpPsiT = Hp @ psi^T  (4x2)
    float HpPsiT[4][2];
    for (int i = 0; i < 4; ++i)
        for (int j = 0; j < 2; ++j) {
            float s = 0.f;
            for (int k = 0; k < 4; ++k) s += Hp[i][k] * psi[j * 4 + k];
            HpPsiT[i][j] = s;
        }
    // S = psi @ HpPsiT + N  (2x2)
    float S[2][2];
    for (int i = 0; i < 2; ++i)
        for (int j = 0; j < 2; ++j) {
            float s = 0.f;
            for (int k = 0; k < 4; ++k) s += psi[i * 4 + k] * HpPsiT[k][j];
            S[i][j] = s + Nmat[i * 2 + j];
        }
    float det = S[0][0] * S[1][1] - S[0][1] * S[1][0];
    float Si[2][2] = {{ S[1][1] / det, -S[0][1] / det},
                      {-S[1][0] / det,  S[0][0] / det}};
    // K = HpPsiT @ Si  (4x2)
    float K[4][2];
    for (int i = 0; i < 4; ++i)
        for (int j = 0; j < 2; ++j)
            K[i][j] = HpPsiT[i][0] * Si[0][j] + HpPsiT[i][1] * Si[1][j];
    // IM = I - K @ psi ; A4 = IM @ phi
    float IM[4][4], A4[4][4];
    for (int i = 0; i < 4; ++i)
        for (int j = 0; j < 4; ++j)
            IM[i][j] = (i == j ? 1.f : 0.f) - (K[i][0] * psi[0 * 4 + j] + K[i][1] * psi[1 * 4 + j]);
    for (int i = 0; i < 4; ++i)
        for (int j = 0; j < 4; ++j) {
            float s = 0.f;
            for (int k = 0; k < 4; ++k) s += IM[i][k] * phi[k * 4 + j];
            A4[i][j] = s;
        }
    // C2 = psi @ A4 (2x4), D2 = psi @ K (2x2)
    float C2[2][4], D2[2][2];
    for (int i = 0; i < 2; ++i)
        for (int j = 0; j < 4; ++j) {
            float s = 0.f;
            for (int k = 0; k < 4; ++k) s += psi[i * 4 + k] * A4[k][j];
            C2[i][j] = s;
        }
    for (int i = 0; i < 2; ++i)
        for (int j = 0; j < 2; ++j) {
            float s = 0.f;
            for (int k = 0; k < 4; ++k) s += psi[i * 4 + k] * K[k][j];
            D2[i][j] = s;
        }
    // H_update = Hp - K @ (psi @ Hp)
    float PsiHp[2][4];
    for (int i = 0; i < 2; ++i)
        for (int j = 0; j < 4; ++j) {
            float s = 0.f;
            for (int k = 0; k < 4; ++k) s += psi[i * 4 + k] * Hp[k][j];
            PsiHp[i][j] = s;
        }
    for (int i = 0; i < 4; ++i)
        for (int j = 0; j < 4; ++j)
            Hout[i * 4 + j] = Hp[i][j] - (K[i][0] * PsiHp[0][j] + K[i][1] * PsiHp[1][j]);

    // Padded combined transform M[16][8]: rows 0-3 = [A4 | K], rows 4-5 = [C2 | D2]
    for (int i = 0; i < 128; ++i) M[i] = 0.f;
    for (int i = 0; i < 4; ++i) {
        for (int j = 0; j < 4; ++j) M[i * 8 + j] = A4[i][j];
        M[i * 8 + 4] = K[i][0];
        M[i * 8 + 5] = K[i][1];
    }
    for (int i = 0; i < 2; ++i) {
        for (int j = 0; j < 4; ++j) M[(4 + i) * 8 + j] = C2[i][j];
        M[(4 + i) * 8 + 4] = D2[i][0];
        M[(4 + i) * 8 + 5] = D2[i][1];
    }
}

// ---------------------------------------------------------------------------
// Streaming kernel: each wave32 processes 16 columns per tile via
// V_WMMA_F32_16X16X4_F32 chained over two K-chunks (z rows, then g rows).
// Bandwidth-bound: all global accesses are 64B-contiguous segments.
// ---------------------------------------------------------------------------
__global__ void kalman_wmma(const float* __restrict__ z,
                            const float* __restrict__ g,
                            const float* __restrict__ M,
                            float* __restrict__ out,
                            long Bc, long ntiles) {
    const int lane = threadIdx.x & 31;
    const int m = lane & 15;               // A row / B column index
    const int k0 = (lane < 16) ? 0 : 2;    // K half held by this lane

    // A-matrix fragments (loop invariant): A1 = M6 cols 0..3, A2 = cols 4..7
    v2f a1, a2;
    a1.x = M[m * 8 + k0];
    a1.y = M[m * 8 + k0 + 1];
    a2.x = M[m * 8 + k0 + 4];
    a2.y = M[m * 8 + k0 + 5];

    float* __restrict__ outg = out;                 // g_update [4,Bc]
    float* __restrict__ outz = out + 4 * Bc + 16;   // z_update [2,Bc] after H_update

    const long wave = ((long)blockIdx.x * blockDim.x + threadIdx.x) >> 5;
    const long nw   = ((long)gridDim.x * blockDim.x) >> 5;

    for (long t = wave; t < ntiles; t += nw) {
        const long col = t * 16 + m;

        // B1: g rows (K=0..3). lanes<16 carry rows 0/1, lanes>=16 rows 2/3.
        v2f b1, b2;
        b1.x = g[(long)k0 * Bc + col];
        b1.y = g[(long)(k0 + 1) * Bc + col];
        // B2: z rows (K=0..1); K=2,3 multiply zero A columns -> upper lanes don't care.
        b2.x = z[col];
        b2.y = z[Bc + col];

        v8f c = {};
        c = __builtin_amdgcn_wmma_f32_16x16x4_f32(false, a2, false, b2,
                                                  (short)0, c, false, false);
        c = __builtin_amdgcn_wmma_f32_16x16x4_f32(false, a1, false, b1,
                                                  (short)0, c, false, false);

        if (lane < 16) {  // D rows 0..5 live in lanes 0..15, VGPRs 0..5
            outg[0 * Bc + col] = c[0];
            outg[1 * Bc + col] = c[1];
            outg[2 * Bc + col] = c[2];
            outg[3 * Bc + col] = c[3];
            outz[0 * Bc + col] = c[4];
            outz[1 * Bc + col] = c[5];
        }
    }
}

// Scalar tail for Bc % 16 != 0 (not hit for B = 4M, kept for generality).
__global__ void kalman_tail(const float* __restrict__ z,
                            const float* __restrict__ g,
                            const float* __restrict__ M,
                            float* __restrict__ out,
                            long Bc, long start) {
    long col = start + (long)blockIdx.x * blockDim.x + threadIdx.x;
    if (col >= Bc) return;
    float x[6] = { g[col], g[Bc + col], g[2 * Bc + col], g[3 * Bc + col],
                   z[col], z[Bc + col] };
    float y[6];
    for (int r = 0; r < 6; ++r) {
        float s = 0.f;
        for (int k = 0; k < 6; ++k) s += M[r * 8 + k] * x[k];
        y[r] = s;
    }
    float* outz = out + 4 * Bc + 16;
    for (int r = 0; r < 4; ++r) out[r * Bc + col] = y[r];
    outz[col] = y[4];
    outz[Bc + col] = y[5];
}

extern "C" void kernel_launch(void* const* d_in, const int* in_sizes, int n_in,
                              void* d_out, int out_size, void* d_ws, size_t ws_size,
                              hipStream_t stream) {
    const float* z   = (const float*)d_in[0];
    const float* N   = (const float*)d_in[1];
    const float* g   = (const float*)d_in[2];
    const float* H   = (const float*)d_in[3];
    const float* phi = (const float*)d_in[4];
    const float* psi = (const float*)d_in[5];
    float* out = (float*)d_out;
    float* M   = (float*)d_ws;   // 128 floats

    const long Bc = (long)in_sizes[0] / 2;

    kalman_setup<<<1, 1, 0, stream>>>(N, H, phi, psi, M, out + 4 * Bc);

    const long ntiles = Bc / 16;
    if (ntiles > 0) {
        const int threads = 256;               // 8 wave32 per block
        long blocksNeeded = (ntiles + 7) / 8;
        int blocks = (int)(blocksNeeded < 32768 ? blocksNeeded : 32768);
        kalman_wmma<<<blocks, threads, 0, stream>>>(z, g, M, out, Bc, ntiles);
    }
    const long rem = Bc - ntiles * 16;
    if (rem > 0) {
        int tb = 256;
        int tblocks = (int)((rem + tb - 1) / tb);
        kalman_tail<<<tblocks, tb, 0, stream>>>(z, g, M, out, Bc, ntiles * 16);
    }
}